// MS_SSIM_L1Loss_36326833389777
// MI455X (gfx1250) — compile-verified
//
#include <hip/hip_runtime.h>
#include <math.h>

typedef __attribute__((ext_vector_type(2))) float v2f;
typedef __attribute__((ext_vector_type(8))) float v8f;

#define TILE 16
#define HALO 16
#define SPAN 48            // TILE + 2*HALO
#define NSIG 5
#define FSZ  33
#define GPAD 80            // padded band row: index d+16, d in [-15,47] -> [1,63]
#define IMG  512
#define NIMG 16
#define KC1   (0.01f * 0.01f)
#define KC2   (0.03f * 0.03f)
#define KALPHA 0.025f
#define KCOMP  20.0f

// fp32 WMMA 16x16x4 : D = A(16x4) * B(4x16) + C, full-precision matrix path on CDNA5
static __device__ __forceinline__ v8f wmma4(v2f a, v2f b, v8f c) {
  return __builtin_amdgcn_wmma_f32_16x16x4_f32(
      /*neg_a=*/false, a, /*neg_b=*/false, b,
      /*c_mod=*/(short)0, c, /*reuse_a=*/false, /*reuse_b=*/false);
}

// Channel value, selected at compile time -> branch-free inner loops.
template <int CH>
static __device__ __forceinline__ float chval(const float* __restrict__ sX,
                                              const float* __restrict__ sY,
                                              int idx) {
  if (CH == 0) return sX[idx];
  if (CH == 1) return sY[idx];
  if (CH == 2) { float t = sX[idx]; return t * t; }
  if (CH == 3) { float t = sY[idx]; return t * t; }
  if (CH == 4) return sX[idx] * sY[idx];
  return fabsf(sX[idx] - sY[idx]);
}

// Separable 33-tap conv of one channel over a 48x48 staged tile -> 16x16 output,
// expressed as banded GEMMs on the fp32 WMMA pipe. The 12 band fragments
// (gband[k-lane]) are shared by all 3 horizontal row-block GEMMs (as B) and by
// the vertical GEMM (as A) -- load them into registers once.
template <int CH>
static __device__ __forceinline__ void conv_pass(
    const float* __restrict__ sX, const float* __restrict__ sY,
    const float* __restrict__ gband,       // gband[d], d in [-15,47], zero-padded
    float* __restrict__ sHw,               // this wave's 48x16 scratch
    float* __restrict__ outp,              // 256-float output tile
    int lm, int hi) {
  // ---- hoist band fragments: bf[kk][v] = gband[kk*4 + v + 2*hi - lm] ----
  v2f bf[12];
  #pragma unroll
  for (int kk = 0; kk < 12; ++kk) {
    #pragma unroll
    for (int v = 0; v < 2; ++v)
      bf[kk][v] = gband[kk * 4 + v + 2 * hi - lm];
  }

  // -------- horizontal pass: H(48x16) = tile(48x48) * Band(48x16) --------
  #pragma unroll
  for (int rb = 0; rb < 3; ++rb) {
    v8f hacc = {};
    const int row = rb * TILE + lm;
    #pragma unroll
    for (int kk = 0; kk < 12; ++kk) {
      v2f a;
      #pragma unroll
      for (int v = 0; v < 2; ++v) {
        int k = kk * 4 + v + 2 * hi;
        a[v] = chval<CH>(sX, sY, row * SPAN + k);   // A[m][k]
      }
      hacc = wmma4(a, bf[kk], hacc);                // B[k][n] = g[k-n]
    }
    #pragma unroll
    for (int v = 0; v < 8; ++v)                     // D row = v + 8*hi
      sHw[(rb * TILE + v + 8 * hi) * TILE + lm] = hacc[v];
  }

  // -------- vertical pass: Out(16x16) = Band^T(16x48) * H(48x16) --------
  v8f vacc = {};
  #pragma unroll
  for (int kk = 0; kk < 12; ++kk) {
    v2f b;
    #pragma unroll
    for (int v = 0; v < 2; ++v) {
      int k = kk * 4 + v + 2 * hi;
      b[v] = sHw[k * TILE + lm];                    // B[k][n]
    }
    vacc = wmma4(bf[kk], b, vacc);                  // A[m][k] = g[k-m]
  }
  #pragma unroll
  for (int v = 0; v < 8; ++v)
    outp[(v + 8 * hi) * TILE + lm] = vacc[v];
}

// Recover separable 1D Gaussians from 2D masks; zero the global accumulator.
__global__ void msssim_prep(const float* __restrict__ gm, float* __restrict__ wsf) {
  int tid = threadIdx.x;
  if (tid == 0) wsf[0] = 0.0f;
  for (int i = tid; i < NSIG * FSZ; i += blockDim.x) {
    int s = i / FSZ, t = i % FSZ;
    const float* m = gm + s * FSZ * FSZ;
    float center = m[16 * FSZ + 16];           // g[16]^2
    wsf[64 + i] = m[16 * FSZ + t] / sqrtf(center);
  }
}

__global__ __launch_bounds__(256)
void msssim_tile(const float* __restrict__ x, const float* __restrict__ y,
                 const float* __restrict__ g1d, float* __restrict__ acc) {
  __shared__ float sX[SPAN * SPAN];      // 9216 B
  __shared__ float sY[SPAN * SPAN];      // 9216 B
  __shared__ float sGpad[NSIG * GPAD];   // zero-padded band rows (no branches on load)
  __shared__ float sH[6][SPAN * TILE];   // per-wave horizontal-pass result (48x16)
  __shared__ float sConv[6][256];        // per-sigma conv outputs (5 chans + l1)
  __shared__ float sProd[256];           // running prod of cs over sigmas
  __shared__ float sLum[256];            // luminance at last sigma
  __shared__ float sRed[8];

  const int tid  = threadIdx.x;
  const int lane = tid & 31;
  const int lm   = lane & 15;            // M (or N) index within 16
  const int hi   = lane >> 4;            // half-wave select
  // Scalarize the wave id so per-channel dispatch is SALU, not exec-mask VALU.
  const int wave = __builtin_amdgcn_readfirstlane(tid >> 5);

  const int tx0 = blockIdx.x * TILE;
  const int ty0 = blockIdx.y * TILE;
  const int img = blockIdx.z;
  const float* xi = x + (size_t)img * IMG * IMG;
  const float* yi = y + (size_t)img * IMG * IMG;

  // ---- stage halo tiles (zero padding outside image, matching conv padding) ----
  for (int i = tid; i < SPAN * SPAN; i += 256) {
    int r = i / SPAN, c = i % SPAN;
    int gr = ty0 - HALO + r, gc = tx0 - HALO + c;
    bool in = (gr >= 0) & (gr < IMG) & (gc >= 0) & (gc < IMG);
    sX[i] = in ? xi[gr * IMG + gc] : 0.0f;
    sY[i] = in ? yi[gr * IMG + gc] : 0.0f;
  }
  // zero-padded band coefficients: sGpad[s][d+16] = g_s[d] for d in [0,32], else 0
  for (int i = tid; i < NSIG * GPAD; i += 256) sGpad[i] = 0.0f;
  __syncthreads();
  for (int i = tid; i < NSIG * FSZ; i += 256) {
    int s = i / FSZ, t = i % FSZ;
    sGpad[s * GPAD + t + 16] = g1d[i];
  }
  sProd[tid] = 1.0f;
  __syncthreads();

  for (int s = 0; s < NSIG; ++s) {
    const float* gband = &sGpad[s * GPAD + 16];
    float* sHw = sH[wave < 6 ? wave : 0];
    // one scalar dispatch per sigma; inner loops are branch-free
    switch (wave) {
      case 0: conv_pass<0>(sX, sY, gband, sHw, sConv[0], lm, hi); break;
      case 1: conv_pass<1>(sX, sY, gband, sHw, sConv[1], lm, hi); break;
      case 2: conv_pass<2>(sX, sY, gband, sHw, sConv[2], lm, hi); break;
      case 3: conv_pass<3>(sX, sY, gband, sHw, sConv[3], lm, hi); break;
      case 4: conv_pass<4>(sX, sY, gband, sHw, sConv[4], lm, hi); break;
      case 5: if (s == NSIG - 1)
                conv_pass<5>(sX, sY, gband, sHw, sConv[5], lm, hi);
              break;
      default: break;                    // waves 6,7 idle during GEMM phase
    }
    __syncthreads();

    // -------- pointwise SSIM terms for this sigma (one pixel per thread) --------
    {
      float mux = sConv[0][tid], muy = sConv[1][tid];
      float cxx = sConv[2][tid], cyy = sConv[3][tid], cxy = sConv[4][tid];
      float mux2 = mux * mux, muy2 = muy * muy, muxy = mux * muy;
      float cs = (2.0f * (cxy - muxy) + KC2) /
                 ((cxx - mux2) + (cyy - muy2) + KC2);
      sProd[tid] *= cs;
      if (s == NSIG - 1)
        sLum[tid] = (2.0f * muxy + KC1) / (mux2 + muy2 + KC1);
    }
    __syncthreads();   // protect sConv before next sigma overwrites it
  }

  // -------- combine + block reduction --------
  float l1g = sConv[5][tid];
  float loss = KALPHA * (1.0f - sLum[tid] * sProd[tid]) + (1.0f - KALPHA) * l1g;

  float v = loss;
  #pragma unroll
  for (int off = 16; off > 0; off >>= 1) v += __shfl_down(v, off, 32);
  if (lane == 0) sRed[wave] = v;
  __syncthreads();
  if (tid == 0) {
    float t = 0.0f;
    #pragma unroll
    for (int i = 0; i < 8; ++i) t += sRed[i];
    atomicAdd(acc, t);
  }
}

__global__ void msssim_finalize(const float* __restrict__ wsf, float* __restrict__ out) {
  out[0] = wsf[0] * (KCOMP / (float)((size_t)NIMG * IMG * IMG));
}

extern "C" void kernel_launch(void* const* d_in, const int* in_sizes, int n_in,
                              void* d_out, int out_size, void* d_ws, size_t ws_size,
                              hipStream_t stream) {
  (void)in_sizes; (void)n_in; (void)out_size; (void)ws_size;
  const float* x  = (const float*)d_in[0];
  const float* y  = (const float*)d_in[1];
  const float* gm = (const float*)d_in[2];
  float* wsf = (float*)d_ws;        // wsf[0] = accumulator, wsf[64..] = 1D filters
  float* out = (float*)d_out;

  msssim_prep<<<1, 256, 0, stream>>>(gm, wsf);
  dim3 grid(IMG / TILE, IMG / TILE, NIMG);
  msssim_tile<<<grid, 256, 0, stream>>>(x, y, wsf + 64, wsf);
  msssim_finalize<<<1, 1, 0, stream>>>(wsf, out);
}